// ProST_init_7404523618738
// MI455X (gfx1250) — compile-verified
//
#include <hip/hip_runtime.h>
#include <math.h>

// ProST / DRR projector for MI455X (gfx1250, wave32).
// Geometry constants from the reference:
static __device__ __forceinline__ float clamp01_127(float v) {
    return fminf(fmaxf(v, 0.0f), 127.0f);
}

typedef float v2f __attribute__((ext_vector_type(2)));
typedef float v8f __attribute__((ext_vector_type(8)));

#define SRC_Z   4.0f
#define DETPLZ  (-2.0f)   // SRC - DET = 4 - 6
#define PIXSP   0.02f
#define MSTEPS  256
#define VOXN    128
#define IMGN    256

// ---------------------------------------------------------------------------
// Kernel 1: ray-distance range from the 8 transformed corner points.
// ws[0] = dmin, ws[1] = dmax. One wave.
// ---------------------------------------------------------------------------
__global__ void prost_setup_kernel(const float* __restrict__ T,
                                   const float* __restrict__ corner,
                                   float* __restrict__ ws) {
    const int l = threadIdx.x;  // 0..31, wave32
    float cx = 0.f, cy = 0.f, cz = 0.f;
    if (l < 8) {
        cx = corner[l * 3 + 0];
        cy = corner[l * 3 + 1];
        cz = corner[l * 3 + 2];
    }
    const float tx = T[0] * cx + T[1] * cy + T[2]  * cz + T[3];
    const float ty = T[4] * cx + T[5] * cy + T[6]  * cz + T[7];
    const float tz = T[8] * cx + T[9] * cy + T[10] * cz + T[11];
    const float dx = tx, dy = ty, dz = tz - SRC_Z;
    const float d  = sqrtf(dx * dx + dy * dy + dz * dz);
    float dmin = (l < 8) ? d :  1e30f;
    float dmax = (l < 8) ? d : -1e30f;
    // xor-reduce within the 8 active lanes (offsets stay inside lanes 0..7)
    #pragma unroll
    for (int off = 4; off >= 1; off >>= 1) {
        dmin = fminf(dmin, __shfl_xor(dmin, off, 32));
        dmax = fmaxf(dmax, __shfl_xor(dmax, off, 32));
    }
    if (l == 0) { ws[0] = dmin; ws[1] = dmax; }
}

// ---------------------------------------------------------------------------
// Kernel 2: main projector.
// Block = (32,8): each wave = 32 contiguous pixels of one detector row.
// Per-wave: two V_WMMA_F32_16X16X4_F32 compute slope = (63.5*T[:, :3]) @ dir
// for all 32 pixels; then a 256-step incremental ray march with trilinear
// gathers from the (L2-resident, 8 MB) volume.
// ---------------------------------------------------------------------------
__launch_bounds__(256)
__global__ void prost_main_kernel(const float* __restrict__ vol,
                                  const float* __restrict__ T,
                                  const float* __restrict__ ws,
                                  float* __restrict__ out) {
    const int lane = threadIdx.x;                 // 0..31
    const int wrow = threadIdx.y;                 // wave id in block, 0..7
    const int w = blockIdx.x * 32 + lane;         // detector column
    const int h = blockIdx.y * 8 + wrow;          // detector row

    // --- per-pixel unit ray direction -------------------------------------
    const float px = ((float)w - 127.5f) * PIXSP;
    const float py = ((float)h - 127.5f) * PIXSP;
    const float rinv = rsqrtf(px * px + py * py + 36.0f);   // |det - src|
    const float dirx = px * rinv;
    const float diry = py * rinv;
    const float dirz = -6.0f * rinv;                         // DETPLZ - SRC_Z

    // --- pack A matrices (16x4 f32, ISA striped layout) -------------------
    // A row m = (dirx_m, diry_m, dirz_m, 0). Lanes 0-15 hold K=0,1 in the two
    // VGPRs; lanes 16-31 hold K=2,3.
    const int srcA  = lane & 15;
    const int srcA2 = (lane & 15) + 16;
    const float s0x = __shfl(dirx, srcA,  32);
    const float s0y = __shfl(diry, srcA,  32);
    const float s0z = __shfl(dirz, srcA,  32);
    const float s1x = __shfl(dirx, srcA2, 32);
    const float s1y = __shfl(diry, srcA2, 32);
    const float s1z = __shfl(dirz, srcA2, 32);
    v2f a1, a2;
    a1.x = (lane < 16) ? s0x : s0z;   // K=0 | K=2
    a1.y = (lane < 16) ? s0y : 0.0f;  // K=1 | K=3 (homogeneous dir.w = 0)
    a2.x = (lane < 16) ? s1x : s1z;
    a2.y = (lane < 16) ? s1y : 0.0f;

    // --- pack B matrix (4x16 f32): B[k][n] = 63.5 * T[n][k] for n<3 -------
    const int n = lane & 15;
    const bool nv = (n < 3);
    const float bk0 = nv ? 63.5f * T[n * 4 + 0] : 0.0f;
    const float bk1 = nv ? 63.5f * T[n * 4 + 1] : 0.0f;
    const float bk2 = nv ? 63.5f * T[n * 4 + 2] : 0.0f;
    v2f bm;
    bm.x = (lane < 16) ? bk0 : bk2;   // K=0 | K=2
    bm.y = (lane < 16) ? bk1 : 0.0f;  // K=1 | K=3

    // --- WMMA: D[m][n] = slope component n of pixel m ---------------------
    v8f czero = {};
    v8f d1 = __builtin_amdgcn_wmma_f32_16x16x4_f32(false, a1, false, bm,
                                                   (short)0, czero, false, false);
    v8f d2 = __builtin_amdgcn_wmma_f32_16x16x4_f32(false, a2, false, bm,
                                                   (short)0, czero, false, false);

    // --- unpack D through per-wave LDS transpose buffers ------------------
    __shared__ float xch[8][2][16][16];           // 16 KB
    {
        const int mbase = (lane >> 4) * 8;        // rows held by this lane
        const int ncol  = lane & 15;
        #pragma unroll
        for (int r = 0; r < 8; ++r) {
            xch[wrow][0][mbase + r][ncol] = d1[r];
            xch[wrow][1][mbase + r][ncol] = d2[r];
        }
    }
    __builtin_amdgcn_wave_barrier();              // same-wave DS ops are in order
    const int which = lane >> 4;
    const int prow  = lane & 15;
    const float slx = xch[wrow][which][prow][0];
    const float sly = xch[wrow][which][prow][1];
    const float slz = xch[wrow][which][prow][2];

    // --- affine ray-march setup ------------------------------------------
    const float dmin = ws[0];
    const float dmax = ws[1];
    const float dt   = (dmax - dmin) * (1.0f / (float)(MSTEPS - 1));
    // base coord b_j = ((T[j][2]*SRC + T[j][3]) * 0.5 + 0.5) * 127
    const float bx = (T[2]  * SRC_Z + T[3])  * 63.5f + 63.5f;
    const float by = (T[6]  * SRC_Z + T[7])  * 63.5f + 63.5f;
    const float bz = (T[10] * SRC_Z + T[11]) * 63.5f + 63.5f;

    float cx = bx + dmin * slx;
    float cy = by + dmin * sly;
    float cz = bz + dmin * slz;
    const float sx = dt * slx;
    const float sy = dt * sly;
    const float sz = dt * slz;

    float acc = 0.0f;

    // 64 outer x 4 inner = 256 samples; one speculative prefetch per outer
    // iteration, ~8 samples ahead of the march front.
    for (int mo = 0; mo < MSTEPS / 4; ++mo) {
        {
            const float pfx = clamp01_127(floorf(cx + 8.0f * sx));
            const float pfy = clamp01_127(floorf(cy + 8.0f * sy));
            const float pfz = clamp01_127(floorf(cz + 8.0f * sz));
            const int pidx = (((int)pfz * VOXN + (int)pfy) * VOXN) + (int)pfx;
            __builtin_prefetch(vol + pidx, 0, 0); // -> global_prefetch_b8
        }
        #pragma unroll
        for (int mi = 0; mi < 4; ++mi) {
            const float x0f = floorf(cx);
            const float y0f = floorf(cy);
            const float z0f = floorf(cz);
            const float wx = cx - x0f;
            const float wy = cy - y0f;
            const float wz = cz - z0f;
            const int ix0 = (int)clamp01_127(x0f);
            const int ix1 = (int)clamp01_127(x0f + 1.0f);
            const int iy0 = (int)clamp01_127(y0f);
            const int iy1 = (int)clamp01_127(y0f + 1.0f);
            const int iz0 = (int)clamp01_127(z0f);
            const int iz1 = (int)clamp01_127(z0f + 1.0f);

            const float* p00 = vol + ((iz0 * VOXN + iy0) << 7);
            const float* p01 = vol + ((iz0 * VOXN + iy1) << 7);
            const float* p10 = vol + ((iz1 * VOXN + iy0) << 7);
            const float* p11 = vol + ((iz1 * VOXN + iy1) << 7);

            const float v000 = p00[ix0], v001 = p00[ix1];
            const float v010 = p01[ix0], v011 = p01[ix1];
            const float v100 = p10[ix0], v101 = p10[ix1];
            const float v110 = p11[ix0], v111 = p11[ix1];

            const float l00 = v000 + wx * (v001 - v000);
            const float l01 = v010 + wx * (v011 - v010);
            const float l10 = v100 + wx * (v101 - v100);
            const float l11 = v110 + wx * (v111 - v110);
            const float l0  = l00 + wy * (l01 - l00);
            const float l1  = l10 + wy * (l11 - l10);
            acc += l0 + wz * (l1 - l0);

            cx += sx; cy += sy; cz += sz;
        }
    }

    out[(h << 8) + w] = acc;
}

// ---------------------------------------------------------------------------
// Host-side launcher (graph-capture safe: only kernel launches on `stream`).
// Inputs (setup_inputs order): x, y(unused), transform_mat3x4, corner_pt.
// ---------------------------------------------------------------------------
extern "C" void kernel_launch(void* const* d_in, const int* in_sizes, int n_in,
                              void* d_out, int out_size, void* d_ws, size_t ws_size,
                              hipStream_t stream) {
    (void)in_sizes; (void)n_in; (void)out_size; (void)ws_size;
    const float* vol    = (const float*)d_in[0];   // (1,1,128,128,128)
    const float* T      = (const float*)d_in[2];   // (1,3,4)
    const float* corner = (const float*)d_in[3];   // (1,8,3)
    float* ws  = (float*)d_ws;
    float* out = (float*)d_out;                    // (1,1,256,256)

    prost_setup_kernel<<<1, 32, 0, stream>>>(T, corner, ws);

    dim3 blk(32, 8);                                // 8 waves / block
    dim3 grd(IMGN / 32, IMGN / 8);                  // (8, 32)
    prost_main_kernel<<<grd, blk, 0, stream>>>(vol, T, ws, out);
}